// GCNLayer_72945724555832
// MI455X (gfx1250) — compile-verified
//
#include <hip/hip_runtime.h>

// GCN layer for MI455X (gfx1250, wave32):
//   node_f = concat(u_f, v_f) @ W        -> WMMA f32 16x16x4 (fp32-faithful, mem-bound anyway)
//   out    = relu(segment_sum(vals * node_f[cols], rows))
//            -> wave-per-edge gather (coalesced 512B row) + global_atomic_add_f32
//               (node_f + out are L2-resident: 51.2MB each vs 192MB L2)

typedef __attribute__((ext_vector_type(2))) float v2f;
typedef __attribute__((ext_vector_type(8))) float v8f;

#define N_USERS 50000
#define N_NODES 100000
#define N_EDGES 1600000
#define D       128
#define MTILES  (N_NODES / 16)      // 6250 exactly
#define LDS_PITCH 132               // pad 128 -> 132 to dodge bank conflicts

// ---------------------------------------------------------------------------
// Kernel 1: node_f[m,n] = sum_k A[m,k] * W[k,n], A = concat(u_f, v_f)
// One wave computes a 16x128 output strip via 8 accumulators (v8f each).
// W is staged transposed into LDS: ldsW[n*PITCH + k] = W[k,n], so a lane's
// B-fragment (two consecutive K values at fixed n) is one 8B LDS load.
// ---------------------------------------------------------------------------
__global__ __launch_bounds__(256) void gemm_wmma_f32(
    const float* __restrict__ u_f, const float* __restrict__ v_f,
    const float* __restrict__ W, float* __restrict__ node_f) {
  __shared__ float ldsW[D * LDS_PITCH];   // 67.6 KB of the 320 KB WGP LDS

  // Cooperative transposed fill (global reads coalesced).
  for (int i = threadIdx.x; i < D * D; i += 256) {
    const int k = i >> 7;
    const int n = i & (D - 1);
    ldsW[n * LDS_PITCH + k] = W[i];
  }
  __syncthreads();

  const int lane  = threadIdx.x & 31;
  const int wave  = threadIdx.x >> 5;
  const int mtile = blockIdx.x * 8 + wave;
  if (mtile >= MTILES) return;          // wave-uniform: EXEC stays all-ones for WMMA
  const int m0 = mtile * 16;

  // A-fragment addressing (32-bit A 16x4 layout):
  //   lanes 0-15  : row m0+lane,    K = kk*4 + {0,1}
  //   lanes 16-31 : row m0+lane-16, K = kk*4 + {2,3}
  const int mrow = m0 + (lane & 15);
  const float* arow = (mrow < N_USERS) ? (u_f + (size_t)mrow * D)
                                       : (v_f + (size_t)(mrow - N_USERS) * D);
  const int khalf = (lane >> 4) << 1;   // 0 or 2

  v8f acc[8] = {};                      // 8 N-tiles of 16 -> 64 accumulator VGPRs

  for (int kk = 0; kk < D / 4; ++kk) {
    const int kb = kk * 4 + khalf;
    const v2f a = *(const v2f*)(arow + kb);      // global_load_b64
#pragma unroll
    for (int nt = 0; nt < 8; ++nt) {
      const int n = nt * 16 + (lane & 15);
      const v2f b = *(const v2f*)(&ldsW[n * LDS_PITCH + kb]);  // ds_load_b64
      acc[nt] = __builtin_amdgcn_wmma_f32_16x16x4_f32(
          false, a, false, b, (short)0, acc[nt], false, false);
    }
  }

  // C/D layout: VGPR r -> row m0+r (lanes 0-15) / m0+r+8 (lanes 16-31)
  const int rbase = m0 + ((lane >> 4) << 3);
  const int ncol  = lane & 15;
#pragma unroll
  for (int r = 0; r < 8; ++r) {
    float* orow = node_f + (size_t)(rbase + r) * D + ncol;
#pragma unroll
    for (int nt = 0; nt < 8; ++nt) orow[nt * 16] = acc[nt][r];
  }
}

// ---------------------------------------------------------------------------
// Kernel 2: zero the accumulator (d_out is poisoned by the harness).
// ---------------------------------------------------------------------------
__global__ __launch_bounds__(256) void zero_f4(float4* __restrict__ out, int n4) {
  const int i = blockIdx.x * 256 + threadIdx.x;
  if (i < n4) {
    float4 z; z.x = 0.f; z.y = 0.f; z.z = 0.f; z.w = 0.f;
    out[i] = z;
  }
}

// ---------------------------------------------------------------------------
// Kernel 3: wave-per-edge scatter-add.
//   32 lanes x float4 = one coalesced 512B row gather of node_f[col]
//   4 non-returning global_atomic_add_f32 per lane into out[row]
// row/col/val forced scalar via readfirstlane -> SGPR base + lane offset.
// ---------------------------------------------------------------------------
__global__ __launch_bounds__(256) void edge_scatter(
    const int* __restrict__ rows, const int* __restrict__ cols,
    const float* __restrict__ vals, const float* __restrict__ node_f,
    float* __restrict__ out) {
  const int e = (blockIdx.x * 256 + threadIdx.x) >> 5;
  if (e >= N_EDGES) return;
  const int lane = threadIdx.x & 31;

  const int   r = __builtin_amdgcn_readfirstlane(rows[e]);
  const int   c = __builtin_amdgcn_readfirstlane(cols[e]);
  const float v = __int_as_float(
      __builtin_amdgcn_readfirstlane(__float_as_int(vals[e])));

  const float4 x = ((const float4*)(node_f + (size_t)c * D))[lane];
  float* dst = out + (size_t)r * D + lane * 4;
  unsafeAtomicAdd(dst + 0, v * x.x);
  unsafeAtomicAdd(dst + 1, v * x.y);
  unsafeAtomicAdd(dst + 2, v * x.z);
  unsafeAtomicAdd(dst + 3, v * x.w);
}

// ---------------------------------------------------------------------------
// Kernel 4: in-place ReLU after all accumulation.
// ---------------------------------------------------------------------------
__global__ __launch_bounds__(256) void relu_f4(float4* __restrict__ out, int n4) {
  const int i = blockIdx.x * 256 + threadIdx.x;
  if (i < n4) {
    float4 x = out[i];
    x.x = fmaxf(x.x, 0.f); x.y = fmaxf(x.y, 0.f);
    x.z = fmaxf(x.z, 0.f); x.w = fmaxf(x.w, 0.f);
    out[i] = x;
  }
}

extern "C" void kernel_launch(void* const* d_in, const int* in_sizes, int n_in,
                              void* d_out, int out_size, void* d_ws, size_t ws_size,
                              hipStream_t stream) {
  const float* u_f  = (const float*)d_in[0];
  const float* v_f  = (const float*)d_in[1];
  const int*   rows = (const int*)  d_in[2];
  const int*   cols = (const int*)  d_in[3];
  const float* vals = (const float*)d_in[4];
  const float* W    = (const float*)d_in[5];
  float* out    = (float*)d_out;
  float* node_f = (float*)d_ws;            // 100000*128*4 = 51.2 MB scratch

  const int n4 = N_NODES * D / 4;          // 3.2M float4

  gemm_wmma_f32<<<(MTILES + 7) / 8, 256, 0, stream>>>(u_f, v_f, W, node_f);
  zero_f4<<<(n4 + 255) / 256, 256, 0, stream>>>((float4*)out, n4);
  edge_scatter<<<(N_EDGES * 32) / 256, 256, 0, stream>>>(rows, cols, vals,
                                                         node_f, out);
  relu_f4<<<(n4 + 255) / 256, 256, 0, stream>>>((float4*)out, n4);
}